// RNNModel_12541304505092
// MI455X (gfx1250) — compile-verified
//
#include <hip/hip_runtime.h>
#include <hip/hip_bf16.h>
#include <math.h>

// ---------------------------------------------------------------------------
// RIM recurrent model for MI455X (gfx1250, wave32).
// GEMMs: V_WMMA_F32_16X16X4_F32 (full fp32 matrix pipe, matches fp32 ref).
// 256-thread blocks stage 128x64 macro-tiles through LDS; full tiles are
// filled with GLOBAL_LOAD_ASYNC_TO_LDS_B128 (ASYNCcnt DMA path), ragged
// edges with predicated sync loads + zero padding.
// ---------------------------------------------------------------------------

#define T_DIM 128
#define BATCH 512
#define NTOK_ 512
#define NINP_ 512
#define NHID_ 600
#define NBLK  6
#define BS_   100
#define TOPK_ 4
#define NH_   4
#define DK_   64
#define DV_   85
#define ATT_  340     // NH_*DV_
#define H2_   4
#define DK2_  32
#define DV2_  32      // H2_*DV2_ = 128

#define KC    32      // K-chunk staged in LDS
#define ASTR  36      // A tile LDS row stride (even -> 8B-aligned b64 frags,
                      // 36 mod 64 pattern -> conflict-free half-wave reads)
#define BSTR  80      // B tile LDS row stride (2*80 mod 64 = 32 -> halves hit
                      // disjoint banks)

typedef float v2f __attribute__((ext_vector_type(2)));
typedef float v8f __attribute__((ext_vector_type(8)));

// ---------------------------------------------------------------------------
// Batched fp32 WMMA GEMM:  C[b] = A[b] (MxK) * B[b] (KxN) + bias
// Call-site invariants here: M % 128 == 0, K % 4 == 0, lda % 4 == 0,
// ldb % 4 == 0, 16B-aligned buffers. N may be ragged (300/340/100).
// Block = 256 threads = 8 waves; block computes a 128x64 C macro-tile;
// wave w owns rows [m0+16w, m0+16w+16) x 64 cols (4 accumulators).
//
// WMMA f32 16x16x4 fragments (wave32):
//   A: lanes 0-15 row m, VGPR0/1 = K0/K1; lanes 16-31 same rows, K2/K3.
//   B: VGPR0 = {K0 | K2}, VGPR1 = {K1 | K3}, col = n0 + (lane&15).
//   C: VGPR r -> row r (lanes 0-15) / row r+8 (lanes 16-31).
// ---------------------------------------------------------------------------
__global__ __launch_bounds__(256) void gemm_wmma_lds(
    const float* __restrict__ A, const float* __restrict__ Bw,
    const float* __restrict__ bias, float* __restrict__ C,
    int M, int N, int K, int lda, int ldb, int ldc,
    long long sA, long long sB, long long sC)
{
    __shared__ float As[128 * ASTR];
    __shared__ float Bs[KC * BSTR];

    const int batch = blockIdx.z;
    A  += (long long)batch * sA;
    Bw += (long long)batch * sB;
    C  += (long long)batch * sC;

    const int m0 = blockIdx.y << 7;   // 128-row macro tile (M % 128 == 0)
    const int n0 = blockIdx.x << 6;   // 64-col macro tile

    const int tid  = threadIdx.x;
    const int wid  = tid >> 5;
    const int lane = tid & 31;
    const int half = lane >> 4;
    const int l16  = lane & 15;

    v8f acc[4] = {{}, {}, {}, {}};

    const bool fullN = (n0 + 64) <= N;

    for (int kc = 0; kc < K; kc += KC) {
        const bool fullK = (kc + KC) <= K;

        if (fullK && fullN) {
            // ---------- async DMA fill: global -> LDS (ASYNCcnt) ----------
            #pragma unroll
            for (int i = 0; i < 4; ++i) {               // A: 1024 float4
                const int e   = tid + (i << 8);
                const int row = e >> 3;
                const int c4  = (e & 7) << 2;
                const unsigned lds_off =
                    (unsigned)(uintptr_t)&As[row * ASTR + c4];
                const float* g = A + (long long)(m0 + row) * lda + kc + c4;
                asm volatile("global_load_async_to_lds_b128 %0, %1, off"
                             :: "v"(lds_off), "v"(g) : "memory");
            }
            #pragma unroll
            for (int i = 0; i < 2; ++i) {               // B: 512 float4
                const int e   = tid + (i << 8);
                const int row = e >> 4;
                const int c4  = (e & 15) << 2;
                const unsigned lds_off =
                    (unsigned)(uintptr_t)&Bs[row * BSTR + c4];
                const float* g = Bw + (long long)(kc + row) * ldb + n0 + c4;
                asm volatile("global_load_async_to_lds_b128 %0, %1, off"
                             :: "v"(lds_off), "v"(g) : "memory");
            }
            asm volatile("s_wait_asynccnt 0x0" ::: "memory");
        } else {
            // ---------- ragged edge: predicated fill, zero padding --------
            #pragma unroll
            for (int i = 0; i < 4; ++i) {
                const int e   = tid + (i << 8);
                const int row = e >> 3;
                const int c4  = (e & 7) << 2;
                const bool ok = (kc + c4) < K;          // K%4==0: whole quad
                const float* g = A + (long long)(m0 + row) * lda + kc + c4;
                float4 v = make_float4(0.f, 0.f, 0.f, 0.f);
                if (ok) v = *(const float4*)g;
                *(float4*)&As[row * ASTR + c4] = v;
            }
            #pragma unroll
            for (int i = 0; i < 2; ++i) {
                const int e     = tid + (i << 8);
                const int row   = e >> 4;
                const int c4    = (e & 15) << 2;
                const bool rowOK = (kc + row) < K;
                const float* g = Bw + (long long)(kc + row) * ldb + n0 + c4;
                #pragma unroll
                for (int q = 0; q < 4; ++q) {
                    const bool ok = rowOK && (n0 + c4 + q) < N;
                    Bs[row * BSTR + c4 + q] = ok ? g[q] : 0.0f;
                }
            }
        }
        __syncthreads();

        // ---------- compute: 8 k-steps x 4 n-subtiles = 32 WMMA ----------
        #pragma unroll
        for (int k4 = 0; k4 < KC; k4 += 4) {
            const int ka = k4 + (half << 1);
            v2f a;                                       // ds_load_b64
            a.x = As[(wid * 16 + l16) * ASTR + ka];
            a.y = As[(wid * 16 + l16) * ASTR + ka + 1];
            #pragma unroll
            for (int j = 0; j < 4; ++j) {
                v2f b;
                b.x = Bs[ka * BSTR + (j << 4) + l16];
                b.y = Bs[(ka + 1) * BSTR + (j << 4) + l16];
                acc[j] = __builtin_amdgcn_wmma_f32_16x16x4_f32(
                    false, a, false, b, (short)0, acc[j], false, false);
            }
        }
        __syncthreads();
    }

    // ---------------- epilogue: predicated only on ragged N ----------------
    const int rbase = m0 + (wid << 4) + (half << 3);
    #pragma unroll
    for (int j = 0; j < 4; ++j) {
        const int col = n0 + (j << 4) + l16;
        if (col < N) {
            const float bv = bias ? bias[col] : 0.0f;
            #pragma unroll
            for (int r = 0; r < 8; ++r)
                C[(long long)(rbase + r) * ldc + col] = acc[j][r] + bv;
        }
    }
}

// --------------------------- elementwise kernels ---------------------------

// h *= mask_t (episode reset gating), in place on the scan state.
__global__ void gate_h_kernel(float* __restrict__ h, const float* __restrict__ mt)
{
    int i = blockIdx.x * blockDim.x + threadIdx.x;
    if (i >= BATCH * NHID_) return;
    h[i] *= mt[i / NHID_];
}

// Null slot has k=v=0, so softmax([a,0]) collapses to sigmoid(a).
__global__ void att_logits_kernel(const float* __restrict__ q,
                                  const float* __restrict__ k,
                                  float* __restrict__ att0)
{
    int idx = blockIdx.x * blockDim.x + threadIdx.x;     // (b, n, h)
    if (idx >= BATCH * NBLK * NH_) return;
    int h  = idx % NH_;
    int bn = idx / NH_;
    int b  = bn / NBLK;
    const float* qrow = q + (long long)bn * (NH_ * DK_) + h * DK_;
    const float* krow = k + (long long)b  * (NH_ * DK_) + h * DK_;
    float s = 0.0f;
    #pragma unroll 8
    for (int d = 0; d < DK_; ++d) s += qrow[d] * krow[d];
    s *= 0.125f;                                          // 1/sqrt(64)
    att0[idx] = 1.0f / (1.0f + expf(-s));
}

// act[b,n] = mean_h att0; mask = (act >= 4th-largest act).
__global__ void act_mask_kernel(const float* __restrict__ att0,
                                float* __restrict__ mblk)
{
    int b = blockIdx.x * blockDim.x + threadIdx.x;
    if (b >= BATCH) return;
    float act[NBLK], tmp[NBLK];
    for (int n = 0; n < NBLK; ++n) {
        const float* a = att0 + ((long long)(b * NBLK + n)) * NH_;
        act[n] = 0.25f * (a[0] + a[1] + a[2] + a[3]);
        tmp[n] = act[n];
    }
    float kth = -1e30f;
    for (int s = 0; s < TOPK_; ++s) {                     // 4th largest of 6
        int best = 0;
        for (int n = 1; n < NBLK; ++n) if (tmp[n] > tmp[best]) best = n;
        kth = tmp[best];
        tmp[best] = -1e30f;
    }
    for (int n = 0; n < NBLK; ++n)
        mblk[b * NBLK + n] = (act[n] >= kth) ? 1.0f : 0.0f;
}

// inp_use[b,n,h*85+dv] = att0[b,n,h] * v[b,h*85+dv]
__global__ void inp_use_kernel(const float* __restrict__ att0,
                               const float* __restrict__ v,
                               float* __restrict__ iuse)
{
    long long idx = (long long)blockIdx.x * blockDim.x + threadIdx.x;
    if (idx >= (long long)BATCH * NBLK * ATT_) return;
    int e  = (int)(idx % ATT_);
    int bn = (int)(idx / ATT_);
    int b  = bn / NBLK;
    int h  = e / DV_;
    iuse[idx] = att0[bn * NH_ + h] * v[(long long)b * ATT_ + e];
}

// Block-diagonal GRU gates. gi/gh are [B,NB,300] (r|z|n), h is gated state.
__global__ void gru_kernel(const float* __restrict__ gi,
                           const float* __restrict__ gh,
                           const float* __restrict__ bih,
                           const float* __restrict__ bhh,
                           const float* __restrict__ h,
                           float* __restrict__ hn)
{
    int idx = blockIdx.x * blockDim.x + threadIdx.x;      // b*600 + n*100 + j
    if (idx >= BATCH * NHID_) return;
    int j  = idx % BS_;
    int bn = idx / BS_;                                   // b*6+n
    int n  = bn % NBLK;
    long long g = (long long)bn * (3 * BS_);
    int nb = n * 3 * BS_;
    float r  = 1.0f / (1.0f + expf(-(gi[g + j]            + bih[nb + j]
                                   + gh[g + j]            + bhh[nb + j])));
    float z  = 1.0f / (1.0f + expf(-(gi[g + BS_ + j]      + bih[nb + BS_ + j]
                                   + gh[g + BS_ + j]      + bhh[nb + BS_ + j])));
    float gn = tanhf((gi[g + 2 * BS_ + j] + bih[nb + 2 * BS_ + j])
               + r * (gh[g + 2 * BS_ + j] + bhh[nb + 2 * BS_ + j]));
    hn[idx] = (1.0f - z) * gn + z * h[idx];
}

// Communication attention across the 6 blocks (per batch, per head).
__global__ void comm_att_kernel(const float* __restrict__ qc,
                                const float* __restrict__ kc,
                                const float* __restrict__ vc,
                                float* __restrict__ cop)
{
    int idx = blockIdx.x * blockDim.x + threadIdx.x;      // (b, h2, n)
    if (idx >= BATCH * H2_ * NBLK) return;
    int n  = idx % NBLK;
    int h2 = (idx / NBLK) % H2_;
    int b  = idx / (NBLK * H2_);
    const float INV_SDK2 = 0.17677669529663687f;          // 1/sqrt(32)
    const float* qrow = qc + ((long long)(b * NBLK + n) * (H2_ * DV2_)) + h2 * DK2_;
    float logit[NBLK];
    float mx = -1e30f;
    for (int m = 0; m < NBLK; ++m) {
        const float* krow = kc + ((long long)(b * NBLK + m) * (H2_ * DV2_)) + h2 * DK2_;
        float s = 0.0f;
        #pragma unroll 8
        for (int d = 0; d < DK2_; ++d) s += qrow[d] * krow[d];
        logit[m] = s * INV_SDK2;
        mx = fmaxf(mx, logit[m]);
    }
    float den = 0.0f;
    for (int m = 0; m < NBLK; ++m) { logit[m] = expf(logit[m] - mx); den += logit[m]; }
    float inv = 1.0f / den;
    float* out = cop + ((long long)(b * NBLK + n) * (H2_ * DV2_)) + h2 * DV2_;
    for (int d = 0; d < DV2_; ++d) {
        float s = 0.0f;
        for (int m = 0; m < NBLK; ++m)
            s += logit[m] * vc[((long long)(b * NBLK + m) * (H2_ * DV2_)) + h2 * DV2_ + d];
        out[d] = s * inv;
    }
}

// hf = mask*(hn + co) + (1-mask)*hb ; write scan state and hs[t].
__global__ void finalize_kernel(const float* __restrict__ hn,
                                const float* __restrict__ co,
                                const float* __restrict__ mblk,
                                float* __restrict__ h,
                                float* __restrict__ hs_t)
{
    int idx = blockIdx.x * blockDim.x + threadIdx.x;      // b*600 + n*100 + j
    if (idx >= BATCH * NHID_) return;
    int j  = idx % BS_;
    int bn = idx / BS_;
    float m  = mblk[bn];
    float hc = hn[idx] + co[(long long)bn * BS_ + j];
    float hf = m * hc + (1.0f - m) * h[idx];
    h[idx]   = hf;
    hs_t[idx] = hf;
}

// ------------------------------- host side ---------------------------------

static inline void launch_gemm(const float* A, const float* Bw, const float* bias,
                               float* C, int M, int N, int K,
                               int lda, int ldb, int ldc,
                               long long sA, long long sB, long long sC,
                               int batch, hipStream_t stream)
{
    dim3 grid((N + 63) / 64, M >> 7, batch);
    gemm_wmma_lds<<<grid, 256, 0, stream>>>(A, Bw, bias, C, M, N, K,
                                            lda, ldb, ldc, sA, sB, sC);
}

extern "C" void kernel_launch(void* const* d_in, const int* in_sizes, int n_in,
                              void* d_out, int out_size, void* d_ws, size_t ws_size,
                              hipStream_t stream)
{
    const float* input = (const float*)d_in[0];   // [T,B,512]
    const float* h0    = (const float*)d_in[1];   // [B,600]
    const float* masks = (const float*)d_in[2];   // [T,B,1]
    const float* enc_W = (const float*)d_in[3];   // [512,512]
    const float* enc_b = (const float*)d_in[4];   // [512]
    const float* Wq_in = (const float*)d_in[5];   // [100,256]
    const float* Wk_in = (const float*)d_in[6];   // [512,256]
    const float* Wv_in = (const float*)d_in[7];   // [512,340]
    const float* W_ih  = (const float*)d_in[8];   // [6,340,300]
    const float* W_hh  = (const float*)d_in[9];   // [6,100,300]
    const float* b_ih  = (const float*)d_in[10];  // [6,300]
    const float* b_hh  = (const float*)d_in[11];  // [6,300]
    const float* Wq_c  = (const float*)d_in[12];  // [100,128]
    const float* Wk_c  = (const float*)d_in[13];  // [100,128]
    const float* Wv_c  = (const float*)d_in[14];  // [100,128]
    const float* Wo_c  = (const float*)d_in[15];  // [128,100]
    const float* dec_W = (const float*)d_in[16];  // [600,512]
    const float* dec_b = (const float*)d_in[17];  // [512]
    float* out = (float*)d_out;                   // [T,B,512]

    // ---- workspace carve (floats; every offset is 16B-aligned) ----
    float* ws    = (float*)d_ws;
    float* emb   = ws;                                        // 128*512*512
    float* hs    = emb  + (long long)T_DIM * BATCH * NINP_;   // 128*512*600
    float* hcur  = hs   + (long long)T_DIM * BATCH * NHID_;   // 512*600
    float* kbuf  = hcur + BATCH * NHID_;                      // 512*256
    float* vbuf  = kbuf + BATCH * NH_ * DK_;                  // 512*340
    float* qbuf  = vbuf + BATCH * ATT_;                       // 3072*256
    float* att0  = qbuf + BATCH * NBLK * NH_ * DK_;           // 512*6*4
    float* mblk  = att0 + BATCH * NBLK * NH_;                 // 512*6
    float* iuse  = mblk + BATCH * NBLK;                       // 3072*340
    float* gibuf = iuse + (long long)BATCH * NBLK * ATT_;     // 3072*300
    float* ghbuf = gibuf + (long long)BATCH * NBLK * 3 * BS_; // 3072*300
    float* hnbuf = ghbuf + (long long)BATCH * NBLK * 3 * BS_; // 512*600
    float* qcb   = hnbuf + BATCH * NHID_;                     // 3072*128
    float* kcb   = qcb + BATCH * NBLK * H2_ * DV2_;
    float* vcb   = kcb + BATCH * NBLK * H2_ * DV2_;
    float* cop   = vcb + BATCH * NBLK * H2_ * DV2_;           // 3072*128
    float* cob   = cop + BATCH * NBLK * H2_ * DV2_;           // 3072*100

    const int TB = T_DIM * BATCH;

    // ---- encoder: emb = input @ enc_W + enc_b   [65536,512]x[512,512] ----
    launch_gemm(input, enc_W, enc_b, emb, TB, NINP_, NTOK_,
                NTOK_, NINP_, NINP_, 0, 0, 0, 1, stream);

    // ---- scan state init ----
    hipMemcpyAsync(hcur, h0, sizeof(float) * BATCH * NHID_,
                   hipMemcpyDeviceToDevice, stream);

    const int EW = 256;
    for (int t = 0; t < T_DIM; ++t) {
        const float* emb_t = emb + (long long)t * BATCH * NINP_;
        const float* mt    = masks + (long long)t * BATCH;
        float* hs_t        = hs + (long long)t * BATCH * NHID_;

        gate_h_kernel<<<(BATCH * NHID_ + EW - 1) / EW, EW, 0, stream>>>(hcur, mt);

        // k = x_t @ Wk_in [512,256]; v = x_t @ Wv_in [512,340]
        launch_gemm(emb_t, Wk_in, nullptr, kbuf, BATCH, NH_ * DK_, NINP_,
                    NINP_, NH_ * DK_, NH_ * DK_, 0, 0, 0, 1, stream);
        launch_gemm(emb_t, Wv_in, nullptr, vbuf, BATCH, ATT_, NINP_,
                    NINP_, ATT_, ATT_, 0, 0, 0, 1, stream);
        // q = hb @ Wq_in : view h as [3072,100]
        launch_gemm(hcur, Wq_in, nullptr, qbuf, BATCH * NBLK, NH_ * DK_, BS_,
                    BS_, NH_ * DK_, NH_ * DK_, 0, 0, 0, 1, stream);

        att_logits_kernel<<<(BATCH * NBLK * NH_ + EW - 1) / EW, EW, 0, stream>>>(qbuf, kbuf, att0);
        act_mask_kernel<<<(BATCH + EW - 1) / EW, EW, 0, stream>>>(att0, mblk);
        inp_use_kernel<<<((long long)BATCH * NBLK * ATT_ + EW - 1) / EW, EW, 0, stream>>>(att0, vbuf, iuse);

        // gi[n] = inp_use[:,n,:] @ W_ih[n]  (batched over 6 blocks)
        launch_gemm(iuse, W_ih, nullptr, gibuf, BATCH, 3 * BS_, ATT_,
                    NBLK * ATT_, 3 * BS_, NBLK * 3 * BS_,
                    ATT_, (long long)ATT_ * 3 * BS_, 3 * BS_, NBLK, stream);
        // gh[n] = hb[:,n,:] @ W_hh[n]
        launch_gemm(hcur, W_hh, nullptr, ghbuf, BATCH, 3 * BS_, BS_,
                    NHID_, 3 * BS_, NBLK * 3 * BS_,
                    BS_, (long long)BS_ * 3 * BS_, 3 * BS_, NBLK, stream);

        gru_kernel<<<(BATCH * NHID_ + EW - 1) / EW, EW, 0, stream>>>(
            gibuf, ghbuf, b_ih, b_hh, hcur, hnbuf);

        // comm attention projections: hn [3072,100] @ [100,128]
        launch_gemm(hnbuf, Wq_c, nullptr, qcb, BATCH * NBLK, H2_ * DV2_, BS_,
                    BS_, H2_ * DV2_, H2_ * DV2_, 0, 0, 0, 1, stream);
        launch_gemm(hnbuf, Wk_c, nullptr, kcb, BATCH * NBLK, H2_ * DV2_, BS_,
                    BS_, H2_ * DV2_, H2_ * DV2_, 0, 0, 0, 1, stream);
        launch_gemm(hnbuf, Wv_c, nullptr, vcb, BATCH * NBLK, H2_ * DV2_, BS_,
                    BS_, H2_ * DV2_, H2_ * DV2_, 0, 0, 0, 1, stream);

        comm_att_kernel<<<(BATCH * H2_ * NBLK + EW - 1) / EW, EW, 0, stream>>>(qcb, kcb, vcb, cop);

        // co = cop [3072,128] @ Wo_c [128,100]
        launch_gemm(cop, Wo_c, nullptr, cob, BATCH * NBLK, BS_, H2_ * DV2_,
                    H2_ * DV2_, BS_, BS_, 0, 0, 0, 1, stream);

        finalize_kernel<<<(BATCH * NHID_ + EW - 1) / EW, EW, 0, stream>>>(
            hnbuf, cob, mblk, hcur, hs_t);
    }

    // ---- decoder: out = hs @ dec_W + dec_b   [65536,600]x[600,512] ----
    launch_gemm(hs, dec_W, dec_b, out, TB, NTOK_, NHID_,
                NHID_, NTOK_, NTOK_, 0, 0, 0, 1, stream);

    (void)in_sizes; (void)n_in; (void)out_size; (void)ws_size;
}